// GaussianKernel_12043088298094
// MI455X (gfx1250) — compile-verified
//
#include <hip/hip_runtime.h>

// RBF kernel map: out[m,n] = exp(-(||x_m||^2 - 2 x_m.w_n + ||w_n||^2)) + b[n]
// M = 262144, N = 128, K = 32, fp32 throughout (memory-bound: ~168 MB traffic
// vs 2.15 GFLOP, ~7.2 us HBM floor at 23.3 TB/s on MI455X).
//
// Per block: 256 threads = 8 wave32. Each wave computes a 16x128 tile via
// V_WMMA_F32_16X16X4_F32 (8 N-tiles x 8 K-steps), with w staged in LDS and
// the exp epilogue fused in registers.

typedef __attribute__((ext_vector_type(2))) float v2f;
typedef __attribute__((ext_vector_type(8))) float v8f;

#define C_DIM   32
#define OUT_DIM 128
#define ROWS_PER_BLOCK 128   // 8 waves * 16 rows

__global__ __launch_bounds__(256) void rbf_wmma_kernel(
    const float* __restrict__ x,   // [M, 32]
    const float* __restrict__ w,   // [32, 128]
    const float* __restrict__ b,   // [128]
    float* __restrict__ out)       // [M, 128]
{
    __shared__ float w_s[C_DIM * OUT_DIM];   // 16 KB
    __shared__ float wsq_s[OUT_DIM];
    __shared__ float b_s[OUT_DIM];

    const int tid = threadIdx.x;

    // ---- Prologue: stage w, ||w||^2, b into LDS (shared by all 8 waves) ----
    {
        const float4* w4  = reinterpret_cast<const float4*>(w);
        float4*       ws4 = reinterpret_cast<float4*>(w_s);
        #pragma unroll
        for (int i = 0; i < 4; ++i)
            ws4[tid + i * 256] = w4[tid + i * 256];   // 4096 floats total
    }
    if (tid < OUT_DIM) {
        float s = 0.f;
        #pragma unroll
        for (int k = 0; k < C_DIM; ++k) {
            float v = w[k * OUT_DIM + tid];
            s += v * v;
        }
        wsq_s[tid] = s;
        b_s[tid]   = b[tid];
    }
    __syncthreads();

    const int lane   = tid & 31;
    const int wave   = tid >> 5;
    const int m_base = blockIdx.x * ROWS_PER_BLOCK + wave * 16;
    const int mrow   = lane & 15;          // A-matrix row held by this lane
    const int koff   = (lane >> 4) * 2;    // lanes 16-31 hold K%4 in {2,3}
    const int ncol   = lane & 15;          // B/C/D column within N-tile
    const int hi8    = (lane >> 4) << 3;   // C/D: lanes 16-31 hold rows M+8

    // ---- Load A: this wave's 16x32 fp32 tile of x, in WMMA A layout ----
    // Lane l, K-step s: holds x[m_base + (l&15)][4s + koff + {0,1}]
    v2f a[8];
    const float* xrow = x + (size_t)(m_base + mrow) * C_DIM;
    #pragma unroll
    for (int s = 0; s < 8; ++s)
        a[s] = *reinterpret_cast<const v2f*>(xrow + 4 * s + koff);

    // ---- ||x||^2 per row: partial per half-lane, combine across halves ----
    float part = 0.f;
    #pragma unroll
    for (int s = 0; s < 8; ++s)
        part += a[s].x * a[s].x + a[s].y * a[s].y;
    float xsq = part + __shfl_xor(part, 16, 32);   // lane l has ||x_{l&15}||^2

    // Broadcast to C/D layout: accum reg r on lane l is row (r + hi8)
    float xs[8];
    #pragma unroll
    for (int r = 0; r < 8; ++r)
        xs[r] = __shfl(xsq, r + hi8, 32);          // ds_bpermute

    // ---- Main: 8 N-tiles x 8 K-steps of V_WMMA_F32_16X16X4_F32 ----
    #pragma unroll
    for (int nt = 0; nt < 8; ++nt) {
        const int n = nt * 16 + ncol;
        v8f acc = {0.f, 0.f, 0.f, 0.f, 0.f, 0.f, 0.f, 0.f};
        #pragma unroll
        for (int s = 0; s < 8; ++s) {
            // B (4x16) layout mirrors A: lane l holds w[4s+koff+{0,1}][n]
            v2f bb;
            bb.x = w_s[(4 * s + koff)     * OUT_DIM + n];
            bb.y = w_s[(4 * s + koff + 1) * OUT_DIM + n];
            acc = __builtin_amdgcn_wmma_f32_16x16x4_f32(
                /*neg_a=*/false, a[s], /*neg_b=*/false, bb,
                /*c_mod=*/(short)0, acc, /*reuse_a=*/false, /*reuse_b=*/false);
        }

        // ---- Fused epilogue: exp(-(xsq - 2*xw + wsq)) + bias ----
        const float wsq  = wsq_s[n];
        const float bias = b_s[n];
        #pragma unroll
        for (int r = 0; r < 8; ++r) {
            float d   = xs[r] - 2.0f * acc[r] + wsq;
            float val = __builtin_amdgcn_exp2f(d * -1.44269504088896340736f) + bias;
            out[(size_t)(m_base + r + hi8) * OUT_DIM + n] = val;
        }
    }
}

extern "C" void kernel_launch(void* const* d_in, const int* in_sizes, int n_in,
                              void* d_out, int out_size, void* d_ws, size_t ws_size,
                              hipStream_t stream) {
    const float* x = (const float*)d_in[0];   // [16*128*128, 32] flat
    const float* w = (const float*)d_in[1];   // [32, 128]
    const float* b = (const float*)d_in[2];   // [128]
    float* out = (float*)d_out;               // [262144, 128]

    const int M = in_sizes[0] / C_DIM;        // 262144
    dim3 grid(M / ROWS_PER_BLOCK);            // 2048 blocks, exact cover
    dim3 block(256);                          // 8 wave32
    rbf_wmma_kernel<<<grid, block, 0, stream>>>(x, w, b, out);
}